// SparseConv3DLayer_11888469476364
// MI455X (gfx1250) — compile-verified
//
#include <hip/hip_runtime.h>

typedef float v2f __attribute__((ext_vector_type(2)));
typedef float v8f __attribute__((ext_vector_type(8)));

constexpr int Bc = 2, Hc = 96, Wc = 128, Dc = 64, Cc = 32, Fc = 32;
// LDS stages one kh-group: 9 taps, interleaved klds[tap][c>>1][f][c&1]
// -> tap*1024 + (c>>1)*64 + f*2 + (c&1).  9 * 1024 floats = 36,864 B static LDS.
constexpr int KGROUP_FLOATS = 9 * (Cc / 2) * Fc * 2;   // 9216

__global__ __launch_bounds__(256) void sparse_conv3d_wmma(
    const float* __restrict__ images,
    const int*   __restrict__ base_plane,
    const float* __restrict__ kern,
    const float* __restrict__ defv_p,
    float*       __restrict__ out)
{
    __shared__ float klds[KGROUP_FLOATS];

    const int tid = threadIdx.x;

    // ---- Per-wave geometry: block = 8 waves = 2 (h,w) columns x 4 depth tiles.
    const int colIdx = blockIdx.x * 2 + (tid >> 7);  // (b*H + h)*W + w
    const int w = colIdx % Wc;
    const int h = (colIdx / Wc) % Hc;
    const int b = colIdx / (Wc * Hc);

    const int lane = tid & 31;
    const int m    = lane & 15;                      // A row / B,D column
    const int hi   = lane >> 4;                      // K-half selector
    const int d0   = ((tid >> 5) & 3) * 16;          // wave's depth tile

    const float defv = defv_p[0];
    const int   bp_c = base_plane[colIdx];

    // Prefetch the 9 neighbor image columns (speculative gfx1250 prefetch).
    for (int kh = 0; kh < 3; ++kh) {
        for (int kw = 0; kw < 3; ++kw) {
            const int hn = h + kh - 1, wn = w + kw - 1;
            if (hn >= 0 && hn < Hc && wn >= 0 && wn < Wc) {
                const size_t nidx = (size_t)(b * Hc + hn) * Wc + wn;
                __builtin_prefetch(images + nidx * (Dc * Cc) + (size_t)(d0 + m) * Cc, 0, 0);
            }
        }
    }

    v8f acc0 = {0.f, 0.f, 0.f, 0.f, 0.f, 0.f, 0.f, 0.f};
    v8f acc1 = acc0;

    for (int kh = 0; kh < 3; ++kh) {
        // ---- Stage this kh-group's 9 taps (36 KB) into LDS, interleaved.
        __syncthreads();                 // previous group fully consumed
        {
            const float* kgrp = kern + (size_t)kh * KGROUP_FLOATS;
            for (int q = tid; q < KGROUP_FLOATS / 4; q += 256) {   // 9 float4 each
                const int gi = q * 4;                 // f-contiguous global index
                const float4 g = *(const float4*)&kgrp[gi];
                const int tap = gi >> 10;             // local tap: kw*3+kd
                const int r   = gi & 1023;
                const int c   = r >> 5;
                const int f   = r & 31;
                float* dst = &klds[(tap * 16 + (c >> 1)) * 64 + f * 2 + (c & 1)];
                dst[0] = g.x; dst[2] = g.y; dst[4] = g.z; dst[6] = g.w;
            }
        }
        __syncthreads();                 // group resident

        const int hn = h + kh - 1;
        for (int kw = 0; kw < 3; ++kw) {
            const int wn = w + kw - 1;
            const bool sp_valid = (hn >= 0) && (hn < Hc) && (wn >= 0) && (wn < Wc);
            int off = 0;
            const float* imgcol = images;
            if (sp_valid) {
                const int nidx = (b * Hc + hn) * Wc + wn;
                off    = bp_c - base_plane[nidx];
                imgcol = images + (size_t)nidx * (Dc * Cc);
            }
            for (int kd = 0; kd < 3; ++kd) {
                const float* tapls = klds + (kw * 3 + kd) * 1024;

                // 16 B fragments: one ds_load_b64 each, immediate offsets.
                v2f bfrag[16];
#pragma unroll
                for (int kc = 0; kc < 8; ++kc) {
                    const float* bp0 = tapls + (kc * 2 + hi) * 64 + m * 2;
                    bfrag[2 * kc]     = *(const v2f*)bp0;         // N = m
                    bfrag[2 * kc + 1] = *(const v2f*)(bp0 + 32);  // N = m+16
                }

                // A fragments: 8 x global_load_b64 off one row pointer.
                const int  shift = off + kd - 1;
                const int  dsrc  = d0 + m + shift;
                const bool rv    = sp_valid && (dsrc >= 0) && (dsrc < Dc);
                const float* rowp = imgcol + (size_t)(rv ? dsrc : 0) * Cc + hi * 2;
                v2f arow[8];
#pragma unroll
                for (int kc = 0; kc < 8; ++kc) {
                    if (rv) {
                        arow[kc] = *(const v2f*)(rowp + kc * 4);
                    } else {
                        v2f z; z.x = defv; z.y = defv;
                        arow[kc] = z;
                    }
                }

                // 16-WMMA burst, two independent accumulator chains.
#pragma unroll
                for (int kc = 0; kc < 8; ++kc) {
                    acc0 = __builtin_amdgcn_wmma_f32_16x16x4_f32(
                        false, arow[kc], false, bfrag[2 * kc],     (short)0, acc0, false, false);
                    acc1 = __builtin_amdgcn_wmma_f32_16x16x4_f32(
                        false, arow[kc], false, bfrag[2 * kc + 1], (short)0, acc1, false, false);
                }
            }
        }
    }

    // D layout: lane -> N=m (acc1: +16); element j -> row M = j + 8*hi.
    float* outp = out + (size_t)colIdx * (Dc * Fc) + (size_t)(d0 + hi * 8) * Fc + m;
#pragma unroll
    for (int j = 0; j < 8; ++j) {
        outp[j * Fc + 0]  = acc0[j];
        outp[j * Fc + 16] = acc1[j];
    }
}

extern "C" void kernel_launch(void* const* d_in, const int* in_sizes, int n_in,
                              void* d_out, int out_size, void* d_ws, size_t ws_size,
                              hipStream_t stream) {
    (void)in_sizes; (void)n_in; (void)out_size; (void)d_ws; (void)ws_size;
    const float* images     = (const float*)d_in[0];
    const int*   base_plane = (const int*)d_in[1];
    const float* kern       = (const float*)d_in[2];
    const float* defv       = (const float*)d_in[3];
    float*       out        = (float*)d_out;

    dim3 grid((Bc * Hc * Wc) / 2);     // 12288 blocks, 2 columns each
    dim3 block(256);                   // 8 waves
    hipLaunchKernelGGL(sparse_conv3d_wmma, grid, block, 0, stream,
                       images, base_plane, kern, defv, out);
}